// ConvDMinBatch_56934086476615
// MI455X (gfx1250) — compile-verified
//
#include <hip/hip_runtime.h>
#include <hip/hip_bf16.h>
#include <math.h>

typedef float v2f __attribute__((ext_vector_type(2)));
typedef float v8f __attribute__((ext_vector_type(8)));

#define NROW 1024
#define FEATD 16
#define DDIM 256

// ---------------------------------------------------------------------------
// Kernel 1: Ms = x @ T via V_WMMA_F32_16X16X4_F32 (fp32 WMMA, K=16 as 4 steps
// of 4), epilogue converts directly to PQ[i,k] = (exp(Ms), exp(-Ms)).
// One wave per 16x16 output tile; 1024 tiles = 128 blocks * 8 waves.
// A layout (16x4 f32): lanes 0-15 hold M=lane, K={0,1} in v0/v1;
//                      lanes 16-31 hold M=lane-16, K={2,3}.
// B layout (4x16 f32): v0 = rows K=0 (lanes 0-15) / K=2 (lanes 16-31),
//                      v1 = rows K=1 / K=3, N = lane%16.
// C/D (16x16 f32): vgpr r, lane l -> M = r + 8*(l>=16), N = l%16.
// ---------------------------------------------------------------------------
__global__ __launch_bounds__(256) void k_gemm_pq(const float* __restrict__ x,
                                                 const float* __restrict__ T,
                                                 float2* __restrict__ PQ) {
  const int lane = threadIdx.x & 31;
  const int wave = threadIdx.x >> 5;
  const int tile = blockIdx.x * 8 + wave;   // 0..1023
  const int iTile = tile >> 4;              // / (DDIM/16)
  const int nTile = tile & 15;
  const int i0 = iTile * 16;
  const int n0 = nTile * 16;
  const int half = lane >> 4;
  const int l16 = lane & 15;

  v8f c = {};
  const float* xrow = x + (i0 + l16) * FEATD;
#pragma unroll
  for (int s = 0; s < 4; ++s) {
    const int kk = 4 * s + 2 * half;
    v2f a, b;
    a.x = xrow[kk];
    a.y = xrow[kk + 1];
    b.x = T[kk * DDIM + n0 + l16];
    b.y = T[(kk + 1) * DDIM + n0 + l16];
    c = __builtin_amdgcn_wmma_f32_16x16x4_f32(false, a, false, b,
                                              (short)0, c, false, false);
  }

  const int col = n0 + l16;
#pragma unroll
  for (int r = 0; r < 8; ++r) {
    const int row = i0 + r + 8 * half;
    const float m = c[r];
    float2 pq;
    pq.x = __expf(m);
    pq.y = __expf(-m);
    PQ[row * DDIM + col] = pq;
  }
}

// ---------------------------------------------------------------------------
// Kernel 2: out_T[i,k] = sum_j min(p_i*q_j, p_j*q_i)  ( == sum_j e^{-|mi-mj|} )
// Block: 256 threads -> 32 rows (i) x 32 cols (k) tile. Lane = k (coalesced,
// bank-conflict-free float2 LDS reads); each thread carries 4 rows in regs.
// j streamed through LDS in chunks of 128 rows (32KB static LDS).
// ---------------------------------------------------------------------------
#define KT 32
#define IT 32
#define JT 128

__global__ __launch_bounds__(256) void k_pair(const float2* __restrict__ PQ,
                                              float* __restrict__ outT) {
  __shared__ float2 spq[JT * KT];           // 32 KB
  const int bk = blockIdx.x & 7;            // DDIM/KT = 8 column tiles
  const int bi = blockIdx.x >> 3;           // NROW/IT = 32 row tiles
  const int k0 = bk * KT;
  const int i0 = bi * IT;
  const int kl = threadIdx.x & 31;          // column within tile
  const int g  = threadIdx.x >> 5;          // row group (4 rows each)

  float pi[4], qi[4], acc[4];
#pragma unroll
  for (int r = 0; r < 4; ++r) {
    const float2 pq = PQ[(i0 + g * 4 + r) * DDIM + k0 + kl];
    pi[r] = pq.x;
    qi[r] = pq.y;
    acc[r] = 0.0f;
  }

  for (int jc = 0; jc < NROW; jc += JT) {
    __syncthreads();
    for (int idx = threadIdx.x; idx < JT * KT; idx += 256) {
      const int j = idx >> 5;
      const int k = idx & 31;
      spq[idx] = PQ[(jc + j) * DDIM + k0 + k];
    }
    __syncthreads();
#pragma unroll 4
    for (int j = 0; j < JT; ++j) {
      const float2 pq = spq[(j << 5) | kl];
#pragma unroll
      for (int r = 0; r < 4; ++r)
        acc[r] += fminf(pi[r] * pq.y, pq.x * qi[r]);
    }
  }

#pragma unroll
  for (int r = 0; r < 4; ++r)
    outT[(i0 + g * 4 + r) * DDIM + k0 + kl] = acc[r];
}

// ---------------------------------------------------------------------------
// Kernel 3: out[i] = sigmoid( x[i,:]·w[0:16] + out_T[i,:]·w[16:272] + b )
// One wave32 per row; shuffle-tree reduction.
// ---------------------------------------------------------------------------
__global__ __launch_bounds__(256) void k_fc(const float* __restrict__ x,
                                            const float* __restrict__ outT,
                                            const float* __restrict__ w,
                                            const float* __restrict__ b,
                                            float* __restrict__ out) {
  const int lane = threadIdx.x & 31;
  const int wave = threadIdx.x >> 5;
  const int i = blockIdx.x * 8 + wave;

  float s = 0.0f;
  if (lane < FEATD) s += x[i * FEATD + lane] * w[lane];
#pragma unroll
  for (int k0 = 0; k0 < DDIM; k0 += 32) {
    const int k = k0 + lane;
    s += outT[i * DDIM + k] * w[FEATD + k];
  }
#pragma unroll
  for (int o = 16; o > 0; o >>= 1)
    s += __shfl_xor(s, o, 32);

  if (lane == 0) {
    const float z = s + b[0];
    out[i] = 1.0f / (1.0f + __expf(-z));
  }
}

// ---------------------------------------------------------------------------
extern "C" void kernel_launch(void* const* d_in, const int* in_sizes, int n_in,
                              void* d_out, int out_size, void* d_ws, size_t ws_size,
                              hipStream_t stream) {
  const float* x    = (const float*)d_in[0];   // [1024, 16]
  const float* T    = (const float*)d_in[1];   // [16, 256]
  const float* fc_w = (const float*)d_in[2];   // [272, 1]
  const float* fc_b = (const float*)d_in[3];   // [1]
  float* out = (float*)d_out;                  // [1024, 1]

  float2* PQ   = (float2*)d_ws;                              // 2 MB
  float*  outT = (float*)((char*)d_ws +
                          (size_t)NROW * DDIM * sizeof(float2)); // 1 MB

  (void)in_sizes; (void)n_in; (void)out_size; (void)ws_size;

  k_gemm_pq<<<128, 256, 0, stream>>>(x, T, PQ);
  k_pair<<<256, 256, 0, stream>>>(PQ, outT);
  k_fc<<<128, 256, 0, stream>>>(x, outT, fc_w, fc_b, out);
}